// Pooling_26594437497310
// MI455X (gfx1250) — compile-verified
//
#include <hip/hip_runtime.h>

// -------------------------------------------------------------------------
// Segment mean-pool (graph readout): out[s,f] = mean over nodes n with
// seg[n]==s of x[n,f].  x: [1M, 128] f32, seg: sorted int64, 256 segments.
//
// Roofline: 512MB of x @ 23.3TB/s => ~22us floor; compute is trivial, so the
// kernel streams x once. The segment reduction is expressed as a matmul with
// a one-hot selector and done on-chip with V_WMMA_F32_16X16X4_F32 (exact f32
// FMA): D[16 segs x 16 feats] += A[16x4 one-hot] x B[4 nodes x 16 feats].
// Main loop: 16 nodes / iteration = 4 chained WMMAs, 8 global b32 loads
// issued up front (staggered loadcnt waits), 4 ds b64 loads for seg ids.
// -------------------------------------------------------------------------

typedef __attribute__((ext_vector_type(2))) float v2f;
typedef __attribute__((ext_vector_type(8))) float v8f;

#define N_SEGS 256
#define D_FEAT 128
#define CHUNK  2048
#define BLOCK  256     // 8 waves of 32; wave w owns feature cols [16w,16w+16)

__device__ __forceinline__ v8f wmma_f32(v2f a, v2f b, v8f c) {
  return __builtin_amdgcn_wmma_f32_16x16x4_f32(
      /*neg_a=*/false, a, /*neg_b=*/false, b,
      /*c_mod=*/(short)0, c, /*reuse_a=*/false, /*reuse_b=*/false);
}

__device__ __forceinline__ void flush_acc(const v8f& c, int base, int half,
                                          int fcol, float* __restrict__ sums) {
#pragma unroll
  for (int r = 0; r < 8; ++r) {
    float v = c[r];
    if (v != 0.0f) {                 // inactive segment rows stay exactly 0
      int s = base + r + 8 * half;   // C: VGPR r -> M=r (lanes 0-15), M=8+r (16-31)
      atomicAdd(&sums[s * D_FEAT + fcol], v);
    }
  }
}

__global__ __launch_bounds__(BLOCK) void seg_sum_wmma(
    const float* __restrict__ x, const long long* __restrict__ seg,
    float* __restrict__ sums, int n_nodes) {
  __shared__ __align__(16) int lseg[CHUNK];

  const int tid  = threadIdx.x;
  const int wave = tid >> 5;         // feature tile 0..7
  const int lane = tid & 31;
  const int chunk_start = blockIdx.x * CHUNK;
  const int chunk_end   = min(chunk_start + CHUNK, n_nodes);
  const int chunk_len   = chunk_end - chunk_start;

  // Stage this chunk's segment ids into LDS
  for (int i = tid; i < CHUNK; i += BLOCK) {
    int g = chunk_start + i;
    if (g >= n_nodes) g = n_nodes - 1;
    lseg[i] = (int)seg[g];
  }
  __syncthreads();
  if (chunk_len <= 0) return;

  const int half = lane >> 4;        // A/B K-half: lanes 0-15 -> K{0,1}, 16-31 -> K{2,3}
  const int m    = lane & 15;        // A row (segment slot) / B,C column
  const int fcol = wave * 16 + m;    // this lane's feature column

  // per-lane base pointer: row (chunk_start + 2*half), column fcol
  const float* __restrict__ xp =
      x + (long long)chunk_start * D_FEAT + (long long)(2 * half) * D_FEAT + fcol;

  v8f c = {0.f, 0.f, 0.f, 0.f, 0.f, 0.f, 0.f, 0.f};
  int base = lseg[0];                // 16-segment window [base, base+16)

  const int jmax = chunk_len & ~15;  // full 16-node super-groups (no range guards)
  for (int j = 0; j < jmax; j += 16) {
    // seg ids for this lane's eight K slots: j + {0,4,8,12} + 2*half + {0,1}
    const int2 sp0 = *(const int2*)&lseg[j +  0 + 2 * half];
    const int2 sp1 = *(const int2*)&lseg[j +  4 + 2 * half];
    const int2 sp2 = *(const int2*)&lseg[j +  8 + 2 * half];
    const int2 sp3 = *(const int2*)&lseg[j + 12 + 2 * half];

    // window check: every delta in [0,16) iff the OR of all deltas < 16
    const unsigned orr =
        (unsigned)(sp0.x - base) | (unsigned)(sp0.y - base) |
        (unsigned)(sp1.x - base) | (unsigned)(sp1.y - base) |
        (unsigned)(sp2.x - base) | (unsigned)(sp2.y - base) |
        (unsigned)(sp3.x - base) | (unsigned)(sp3.y - base);

    if (__all(orr < 16u)) {
      // issue all 8 x-loads up front; compiler staggers loadcnt waits
      v2f b0, b1, b2, b3;
      b0.x = xp[(j +  0) * D_FEAT];  b0.y = xp[(j +  0) * D_FEAT + D_FEAT];
      b1.x = xp[(j +  4) * D_FEAT];  b1.y = xp[(j +  4) * D_FEAT + D_FEAT];
      b2.x = xp[(j +  8) * D_FEAT];  b2.y = xp[(j +  8) * D_FEAT + D_FEAT];
      b3.x = xp[(j + 12) * D_FEAT];  b3.y = xp[(j + 12) * D_FEAT + D_FEAT];
      __builtin_prefetch(&xp[(j + 32) * D_FEAT], 0, 1);

      v2f a0, a1, a2, a3;
      a0.x = (sp0.x - base == m) ? 1.0f : 0.0f;
      a0.y = (sp0.y - base == m) ? 1.0f : 0.0f;
      a1.x = (sp1.x - base == m) ? 1.0f : 0.0f;
      a1.y = (sp1.y - base == m) ? 1.0f : 0.0f;
      a2.x = (sp2.x - base == m) ? 1.0f : 0.0f;
      a2.y = (sp2.y - base == m) ? 1.0f : 0.0f;
      a3.x = (sp3.x - base == m) ? 1.0f : 0.0f;
      a3.y = (sp3.y - base == m) ? 1.0f : 0.0f;

      c = wmma_f32(a0, b0, c);
      c = wmma_f32(a1, b1, c);
      c = wmma_f32(a2, b2, c);
      c = wmma_f32(a3, b3, c);
    } else {
      // rare: segment-window boundary (or pathological id gap).
      // Flush, add these 16 nodes via scalar atomics, re-base.
      flush_acc(c, base, half, fcol, sums);
      c = (v8f){0.f, 0.f, 0.f, 0.f, 0.f, 0.f, 0.f, 0.f};
      if (lane < 16) {
#pragma unroll
        for (int k = 0; k < 16; ++k) {
          const int s = lseg[j + k];
          atomicAdd(&sums[s * D_FEAT + wave * 16 + lane],
                    x[(long long)(chunk_start + j + k) * D_FEAT + wave * 16 + lane]);
        }
      }
      const int nb = j + 16;
      base = lseg[nb < chunk_len ? nb : (chunk_len - 1)];
    }
  }
  flush_acc(c, base, half, fcol, sums);

  // ragged tail (< 16 nodes, only the grid's last chunk): scalar path
  if (lane < 16) {
    for (int idx = jmax; idx < chunk_len; ++idx) {
      const int s = lseg[idx];
      atomicAdd(&sums[s * D_FEAT + wave * 16 + lane],
                x[(long long)(chunk_start + idx) * D_FEAT + wave * 16 + lane]);
    }
  }
}

// counts[s] = #nodes with seg==s, via binary search over the sorted ids
__global__ void seg_count(const long long* __restrict__ seg, int n,
                          int* __restrict__ counts) {
  const int s = blockIdx.x * blockDim.x + threadIdx.x;
  if (s >= N_SEGS) return;
  int lo0 = 0, hi0 = n;
  while (lo0 < hi0) { int mid = (lo0 + hi0) >> 1; if (seg[mid] < (long long)s) lo0 = mid + 1; else hi0 = mid; }
  int lo1 = 0, hi1 = n;
  while (lo1 < hi1) { int mid = (lo1 + hi1) >> 1; if (seg[mid] < (long long)(s + 1)) lo1 = mid + 1; else hi1 = mid; }
  counts[s] = lo1 - lo0;
}

__global__ void zero_sums(float* __restrict__ sums) {
  const int i = blockIdx.x * blockDim.x + threadIdx.x;
  if (i < N_SEGS * D_FEAT) sums[i] = 0.0f;
}

__global__ void finalize(const float* __restrict__ sums,
                         const int* __restrict__ counts,
                         float* __restrict__ out) {
  const int i = blockIdx.x * blockDim.x + threadIdx.x;
  if (i < N_SEGS * D_FEAT) {
    const float cnt = (float)counts[i / D_FEAT];
    out[i] = sums[i] / fmaxf(cnt, 1.0f);
  }
}

extern "C" void kernel_launch(void* const* d_in, const int* in_sizes, int n_in,
                              void* d_out, int out_size, void* d_ws, size_t ws_size,
                              hipStream_t stream) {
  const float* x       = (const float*)d_in[0];
  const long long* seg = (const long long*)d_in[1];
  const int n_nodes    = in_sizes[0] / D_FEAT;   // 1,000,000

  float* sums = (float*)d_ws;                    // 256*128 f32 partial sums
  int* counts = (int*)((char*)d_ws + (size_t)N_SEGS * D_FEAT * sizeof(float));

  zero_sums<<<(N_SEGS * D_FEAT + 255) / 256, 256, 0, stream>>>(sums);
  seg_count<<<1, N_SEGS, 0, stream>>>(seg, n_nodes, counts);

  const int nblocks = (n_nodes + CHUNK - 1) / CHUNK;
  seg_sum_wmma<<<nblocks, BLOCK, 0, stream>>>(x, seg, sums, n_nodes);

  finalize<<<(N_SEGS * D_FEAT + 255) / 256, 256, 0, stream>>>(sums, counts,
                                                              (float*)d_out);
}